// Model_69638599737668
// MI455X (gfx1250) — compile-verified
//
#include <hip/hip_runtime.h>

// Problem constants (match reference)
#define B_ 1024
#define D_ 512
#define H_ 256
#define O_ 5
#define E_ 64

typedef float v2f __attribute__((ext_vector_type(2)));
typedef float v8f __attribute__((ext_vector_type(8)));

// ---------------------------------------------------------------------------
// Kernel 1: bucket samples by expert (histogram + prefix + scatter) and zero
// the output (needed: kernel 2 accumulates with global atomic f32 adds).
// B=1024, E=64 -> trivial, single workgroup.
// ---------------------------------------------------------------------------
__global__ __launch_bounds__(256) void bucket_kernel(
    const int* __restrict__ eid,
    int* __restrict__ counts, int* __restrict__ offs,
    int* __restrict__ perm, float* __restrict__ out) {
  __shared__ int cnt[E_];
  __shared__ int off[E_];
  const int t = threadIdx.x;
  if (t < E_) cnt[t] = 0;
  __syncthreads();

  int r[B_ / 256];
  int ii = 0;
  for (int i = t; i < B_; i += 256) r[ii++] = atomicAdd(&cnt[eid[i]], 1);
  for (int i = t; i < B_ * O_; i += 256) out[i] = 0.0f;
  __syncthreads();

  if (t == 0) {
    int s = 0;
    for (int e = 0; e < E_; ++e) { off[e] = s; s += cnt[e]; }
  }
  __syncthreads();

  if (t < E_) { counts[t] = cnt[t]; offs[t] = off[t]; }
  ii = 0;
  for (int i = t; i < B_; i += 256) perm[off[eid[i]] + r[ii++]] = i;
}

// ---------------------------------------------------------------------------
// Kernel 2: grouped GEMM. Block (e, y): expert e, hidden-columns half y
// (128 of 256). 128 threads = 4 wave32; each wave owns 32 hidden cols
// (two 16x16 WMMA N-tiles). Row tiles of M=16 samples.
// fp32 WMMA: V_WMMA_F32_16X16X4_F32, K stepped by 4 over D=512.
// ---------------------------------------------------------------------------
__global__ __launch_bounds__(128) void moe_grouped_gemm_kernel(
    const float* __restrict__ x,  const float* __restrict__ W1,
    const float* __restrict__ b1, const float* __restrict__ W2,
    const float* __restrict__ b2,
    const int* __restrict__ counts, const int* __restrict__ offs,
    const int* __restrict__ perm,   float* __restrict__ out) {
  const int e     = blockIdx.x;
  const int yblk  = blockIdx.y;           // 0 or 1: hidden-column half
  const int tid   = threadIdx.x;
  const int lane  = tid & 31;             // wave32
  const int wave  = tid >> 5;             // 0..3
  const int m     = lane & 15;            // M (A) / N (B,C) index within tile
  const int khalf = (lane >> 4) << 1;     // lanes 16-31 handle K+2,K+3

  const int count = counts[e];
  const int off   = offs[e];
  const float* W1p = W1 + (size_t)e * D_ * H_;
  const float* W2p = W2 + (size_t)e * H_ * O_;

  const int hbase = yblk * 128;           // this block's hidden-col range
  const int n0    = hbase + wave * 32;    // this wave's first column

  // Stride 516 (%64 == 4): conflict-free ds_load_b64 of A fragments.
  __shared__ float Xs[16 * 516];
  // Stride 132 (%64 == 4): conflict-free stores/reads of h tile.
  __shared__ float Hs[16 * 132];
  __shared__ int   smp[16];

  for (int tile = 0; tile * 16 < count; ++tile) {
    if (tid < 16)
      smp[tid] = (tile * 16 + tid < count) ? perm[off + tile * 16 + tid] : -1;
    __syncthreads();

    // Stage x tile (16 rows x 512) into padded LDS; zero-fill tail rows.
    for (int idx = tid; idx < 16 * 128; idx += 128) {
      const int row = idx >> 7;
      const int c4  = (idx & 127) << 2;
      float4 v = make_float4(0.f, 0.f, 0.f, 0.f);
      const int s = smp[row];
      if (s >= 0) v = *(const float4*)(x + (size_t)s * D_ + c4);
      *(float4*)(&Xs[row * 516 + c4]) = v;
    }
    __syncthreads();

    v8f c0 = {};
    v8f c1 = {};
    const float* Bp0 = W1p + n0 + m;      // N-tile 0 column base
    const float* Bp1 = Bp0 + 16;          // N-tile 1

    #pragma unroll 4
    for (int k = 0; k < D_; k += 4) {
      const int kk = k + khalf;
      // A fragment: lane (m, khalf) -> X[m][kk], X[m][kk+1]  (ds_load_b64)
      v2f a;
      a.x = Xs[m * 516 + kk];
      a.y = Xs[m * 516 + kk + 1];
      // B fragments: lane -> W1[kk][n], W1[kk+1][n]
      v2f bb0, bb1;
      bb0.x = Bp0[(size_t)kk * H_];
      bb0.y = Bp0[(size_t)(kk + 1) * H_];
      bb1.x = Bp1[(size_t)kk * H_];
      bb1.y = Bp1[(size_t)(kk + 1) * H_];
      c0 = __builtin_amdgcn_wmma_f32_16x16x4_f32(
          false, a, false, bb0, (short)0, c0, false, false);
      c1 = __builtin_amdgcn_wmma_f32_16x16x4_f32(
          false, a, false, bb1, (short)0, c1, false, false);
    }

    // Bias + ReLU in accumulator; C layout: col = n0 + m constant per lane,
    // row = v + 8*(lane>=16).
    const int   lc    = wave * 32 + m;     // local hidden col within half
    const float bias0 = b1[e * H_ + n0 + m];
    const float bias1 = b1[e * H_ + n0 + 16 + m];
    const int   rbase = (lane >> 4) << 3;  // 0 or 8
    #pragma unroll
    for (int v = 0; v < 8; ++v) {
      float h0 = c0[v] + bias0; h0 = h0 > 0.f ? h0 : 0.f;
      float h1 = c1[v] + bias1; h1 = h1 > 0.f ? h1 : 0.f;
      Hs[(rbase + v) * 132 + lc]      = h0;
      Hs[(rbase + v) * 132 + lc + 16] = h1;
    }
    __syncthreads();

    // Second GEMM: [16 x 128] x [128 x 5]; partial-K sums combined across
    // the two H-half blocks via global atomic f32 add.
    if (tid < 16 * O_) {
      const int row = tid / O_;
      const int col = tid % O_;
      const int s   = smp[row];
      if (s >= 0) {
        float acc = (yblk == 0) ? b2[e * O_ + col] : 0.f;
        #pragma unroll 4
        for (int k = 0; k < 128; ++k)
          acc += Hs[row * 132 + k] * W2p[(size_t)(hbase + k) * O_ + col];
        atomicAdd(out + (size_t)s * O_ + col, acc);
      }
    }
    __syncthreads();   // protect Xs/Hs before next tile
  }
}

// ---------------------------------------------------------------------------
extern "C" void kernel_launch(void* const* d_in, const int* in_sizes, int n_in,
                              void* d_out, int out_size, void* d_ws, size_t ws_size,
                              hipStream_t stream) {
  const float* x   = (const float*)d_in[0];
  const float* W1  = (const float*)d_in[1];
  const float* b1  = (const float*)d_in[2];
  const float* W2  = (const float*)d_in[3];
  const float* b2  = (const float*)d_in[4];
  const int*   eid = (const int*)d_in[5];
  float* out = (float*)d_out;

  int* ws     = (int*)d_ws;
  int* counts = ws;            // [64]
  int* offs   = ws + 64;       // [64]
  int* perm   = ws + 128;      // [1024]

  bucket_kernel<<<1, 256, 0, stream>>>(eid, counts, offs, perm, out);

  dim3 grid(E_, 2);
  moe_grouped_gemm_kernel<<<grid, 128, 0, stream>>>(
      x, W1, b1, W2, b2, counts, offs, perm, out);
}